// GAT_11605001634559
// MI455X (gfx1250) — compile-verified
//
#include <hip/hip_runtime.h>
#include <hip/hip_bf16.h>

#define NNODES 50000
#define NEDGES 800000
#define NDIM   128
#define EDIM   32
#define HIDD   32
#define H1N    4
#define OUTD   64
#define SLOPE  0.2f

typedef __attribute__((ext_vector_type(2))) float v2f;
typedef __attribute__((ext_vector_type(8))) float v8f;

// ---------------- order-preserving float<->uint for atomic segment-max -----
__device__ __forceinline__ unsigned f2ord(float f) {
  unsigned u = __float_as_uint(f);
  return (u & 0x80000000u) ? ~u : (u | 0x80000000u);
}
__device__ __forceinline__ float ord2f(unsigned u) {
  return (u & 0x80000000u) ? __uint_as_float(u & 0x7FFFFFFFu) : __uint_as_float(~u);
}

// ---------------- f32 WMMA GEMM: C[M,N] = op(A)[M,K] * B[K,N], N = NT*16 ----
// One wave per 16-row strip of C: the A fragment is loaded ONCE per k-step and
// reused across all NT output tiles (A HBM traffic 1x instead of NT x).
// V_WMMA_F32_16X16X4_F32 fragment layouts (ISA 7.12.2):
//   A (16x4):  lanes 0-15 hold K=k0,k0+1 ; lanes 16-31 hold K=k0+2,k0+3
//   B (4x16):  VGPR v: lanes 0-15 -> row k0+v ; lanes 16-31 -> row k0+2+v
//   D (16x16): VGPR r: lanes 0-15 -> M=r ; lanes 16-31 -> M=r+8 ; N = lane&15
template<bool RELU_A, int NT>
__global__ __launch_bounds__(256)
void gemm16_wmma_f32(const float* __restrict__ A, const float* __restrict__ B,
                     float* __restrict__ C, int M, int K) {
  const int N = NT * 16;
  const int strip = blockIdx.x * (blockDim.x >> 5) + (threadIdx.x >> 5);
  const int mtiles = M >> 4;             // M = 50000 is a multiple of 16
  if (strip >= mtiles) return;           // wave-uniform: EXEC stays all-ones
  const int lane = threadIdx.x & 31;
  const int half = lane >> 4, idx = lane & 15;
  const int arow = strip * 16 + idx;     // M index of this lane's A row

  v8f acc[NT];
#pragma unroll
  for (int t = 0; t < NT; ++t) acc[t] = (v8f){};

  for (int k0 = 0; k0 < K; k0 += 4) {
    const int ka = k0 + 2 * half;
    float a0 = A[(size_t)arow * K + ka];
    float a1 = A[(size_t)arow * K + ka + 1];
    if (RELU_A) { a0 = fmaxf(a0, 0.f); a1 = fmaxf(a1, 0.f); }
    v2f a; a.x = a0; a.y = a1;
#pragma unroll
    for (int t = 0; t < NT; ++t) {
      const int bcol = t * 16 + idx;
      v2f b; b.x = B[(size_t)ka * N + bcol];
             b.y = B[(size_t)(ka + 1) * N + bcol];
      acc[t] = __builtin_amdgcn_wmma_f32_16x16x4_f32(false, a, false, b,
                                                     (short)0, acc[t], false, false);
    }
  }
#pragma unroll
  for (int t = 0; t < NT; ++t)
#pragma unroll
    for (int r = 0; r < 8; ++r)
      C[(size_t)(strip * 16 + r + 8 * half) * N + t * 16 + idx] = acc[t][r];
}

// -------- eh[n,h] = feat[n,h,:].ah[h,:],  et[n,h] = feat[n,h,:].at[h,:] -----
__global__ void node_attn_kernel(const float* __restrict__ feat,
                                 const float* __restrict__ ah,
                                 const float* __restrict__ at,
                                 float* __restrict__ eh, float* __restrict__ et,
                                 int H, int D) {
  int t = blockIdx.x * blockDim.x + threadIdx.x;
  if (t >= NNODES * H) return;
  int n = t / H, h = t % H;
  const float* f = feat + (size_t)n * H * D + (size_t)h * D;
  float s1 = 0.f, s2 = 0.f;
  for (int d = 0; d < D; ++d) { s1 += f[d] * ah[h * D + d]; s2 += f[d] * at[h * D + d]; }
  eh[t] = s1; et[t] = s2;
}

// -------- ew[k,h] = sum_d We[k, h*D+d] * ae[h,d]   (collapses edge GEMM) ----
__global__ void fold_we_kernel(const float* __restrict__ We,
                               const float* __restrict__ ae,
                               float* __restrict__ ew, int H, int D) {
  int t = blockIdx.x * blockDim.x + threadIdx.x;
  if (t >= EDIM * H) return;
  int k = t / H, h = t % H;
  float s = 0.f;
  for (int d = 0; d < D; ++d) s += We[(size_t)k * H * D + h * D + d] * ae[h * D + d];
  ew[t] = s;
}

// -------- init helpers ------------------------------------------------------
__global__ void init_out_kernel(float* __restrict__ out, const float* __restrict__ b,
                                int HD, long total) {
  long t = (long)blockIdx.x * blockDim.x + threadIdx.x;
  if (t < total) out[t] = b[t % HD];
}
__global__ void init_ms_kernel(unsigned* __restrict__ maxb, float* __restrict__ sumb,
                               int count) {
  int t = blockIdx.x * blockDim.x + threadIdx.x;
  if (t < count) { maxb[t] = 0u; sumb[t] = 0.f; }
}

// -------- per-edge logits: leakyrelu(eh[src]+eattr.ew+et[dst]); seg-max -----
__global__ void edge_logits_kernel(const float* __restrict__ eattr,
                                   const int* __restrict__ src,
                                   const int* __restrict__ dst,
                                   const float* __restrict__ ew,   // [EDIM,H]
                                   const float* __restrict__ eh,
                                   const float* __restrict__ et,
                                   float* __restrict__ logits,
                                   unsigned* __restrict__ maxb, int H) {
  int e = blockIdx.x * blockDim.x + threadIdx.x;
  if (e >= NEDGES) return;
  float er[EDIM];
  const float4* row = (const float4*)(eattr + (size_t)e * EDIM);
#pragma unroll
  for (int q = 0; q < EDIM / 4; ++q) {
    float4 v = row[q];
    er[4 * q] = v.x; er[4 * q + 1] = v.y; er[4 * q + 2] = v.z; er[4 * q + 3] = v.w;
  }
  int s = src[e], d = dst[e];
  for (int h = 0; h < H; ++h) {
    float ee = 0.f;
#pragma unroll
    for (int k = 0; k < EDIM; ++k) ee += er[k] * ew[k * H + h];
    float l = eh[s * H + h] + ee + et[d * H + h];
    l = (l > 0.f) ? l : SLOPE * l;
    logits[(size_t)e * H + h] = l;
    atomicMax(&maxb[d * H + h], f2ord(l));
  }
}

// -------- ex = exp(l - m[dst]); seg-sum (ex overwrites logits in place) -----
__global__ void edge_exp_sum_kernel(const int* __restrict__ dst,
                                    const unsigned* __restrict__ maxb,
                                    float* __restrict__ logits,
                                    float* __restrict__ sumb, int H) {
  long t = (long)blockIdx.x * blockDim.x + threadIdx.x;
  if (t >= (long)NEDGES * H) return;
  int e = (int)(t / H), h = (int)(t % H);
  int d = dst[e];
  float ex = expf(logits[t] - ord2f(maxb[d * H + h]));
  logits[t] = ex;
  atomicAdd(&sumb[d * H + h], ex);
}

// -------- out[dst] += feat[src] * (ex / sum[dst]) ---------------------------
__global__ void edge_agg_kernel(const int* __restrict__ src,
                                const int* __restrict__ dst,
                                const float* __restrict__ feat,
                                const float* __restrict__ ex,
                                const float* __restrict__ sumb,
                                float* __restrict__ out, int H, int D) {
  const int HD = H * D;
  long t = (long)blockIdx.x * blockDim.x + threadIdx.x;
  if (t >= (long)NEDGES * HD) return;
  int e = (int)(t / HD), j = (int)(t % HD);
  int h = j / D;
  int s = src[e], d = dst[e];
  float a = ex[(size_t)e * H + h] / sumb[d * H + h];
  atomicAdd(&out[(size_t)d * HD + j], feat[(size_t)s * HD + j] * a);
}

// ---------------------------------------------------------------------------
static inline int cdiv(long a, long b) { return (int)((a + b - 1) / b); }

extern "C" void kernel_launch(void* const* d_in, const int* in_sizes, int n_in,
                              void* d_out, int out_size, void* d_ws, size_t ws_size,
                              hipStream_t stream) {
  const float* x     = (const float*)d_in[0];
  const float* eattr = (const float*)d_in[1];
  const int*   src   = (const int*)d_in[2];
  const int*   dst   = (const int*)d_in[3];
  const float* W0  = (const float*)d_in[4];
  const float* We0 = (const float*)d_in[5];
  const float* ah0 = (const float*)d_in[6];
  const float* ae0 = (const float*)d_in[7];
  const float* at0 = (const float*)d_in[8];
  const float* b0  = (const float*)d_in[9];
  const float* W1  = (const float*)d_in[10];
  const float* We1 = (const float*)d_in[11];
  const float* ah1 = (const float*)d_in[12];
  const float* ae1 = (const float*)d_in[13];
  const float* at1 = (const float*)d_in[14];
  const float* b1  = (const float*)d_in[15];
  float* out = (float*)d_out;

  // workspace carve-up (all f32/u32)
  float* ws = (float*)d_ws;
  size_t o = 0;
  float* feat0 = ws + o; o += (size_t)NNODES * (H1N * HIDD);   // [N,128]
  float* hbuf  = ws + o; o += (size_t)NNODES * (H1N * HIDD);   // [N,128] layer0 out
  float* feat1 = ws + o; o += (size_t)NNODES * OUTD;           // [N,64]
  float* ehb   = ws + o; o += (size_t)NNODES * H1N;
  float* etb   = ws + o; o += (size_t)NNODES * H1N;
  float* ewb   = ws + o; o += (size_t)EDIM * H1N;
  float* exb   = ws + o; o += (size_t)NEDGES * H1N;            // logits -> ex
  unsigned* maxb = (unsigned*)(ws + o); o += (size_t)NNODES * H1N;
  float* sumb  = ws + o; o += (size_t)NNODES * H1N;

  const int strips = NNODES / 16;                 // 3125 row strips
  const int gemm_blocks = cdiv((long)strips * 32, 256);

  // ===================== layer 0 : H=4, D=32, HD=128 =======================
  {
    const int H = H1N, D = HIDD, HD = H * D;
    // feat0 = x @ W0   (50000x128 @ 128x128), f32 WMMA, 8 N-tiles per wave
    gemm16_wmma_f32<false, 8><<<gemm_blocks, 256, 0, stream>>>(
        x, W0, feat0, NNODES, NDIM);
    node_attn_kernel<<<cdiv((long)NNODES * H, 256), 256, 0, stream>>>(
        feat0, ah0, at0, ehb, etb, H, D);
    fold_we_kernel<<<1, 256, 0, stream>>>(We0, ae0, ewb, H, D);
    init_ms_kernel<<<cdiv((long)NNODES * H, 256), 256, 0, stream>>>(
        maxb, sumb, NNODES * H);
    edge_logits_kernel<<<cdiv(NEDGES, 256), 256, 0, stream>>>(
        eattr, src, dst, ewb, ehb, etb, exb, maxb, H);
    edge_exp_sum_kernel<<<cdiv((long)NEDGES * H, 256), 256, 0, stream>>>(
        dst, maxb, exb, sumb, H);
    init_out_kernel<<<cdiv((long)NNODES * HD, 256), 256, 0, stream>>>(
        hbuf, b0, HD, (long)NNODES * HD);
    edge_agg_kernel<<<cdiv((long)NEDGES * HD, 256), 256, 0, stream>>>(
        src, dst, feat0, exb, sumb, hbuf, H, D);
    // ReLU on hbuf is fused into the layer-1 GEMM A-load below.
  }

  // ===================== layer 1 : H=1, D=64, HD=64 ========================
  {
    const int H = 1, D = OUTD, HD = H * D;
    // feat1 = relu(hbuf) @ W1   (50000x128 @ 128x64), 4 N-tiles per wave
    gemm16_wmma_f32<true, 4><<<gemm_blocks, 256, 0, stream>>>(
        hbuf, W1, feat1, NNODES, H1N * HIDD);
    node_attn_kernel<<<cdiv((long)NNODES * H, 256), 256, 0, stream>>>(
        feat1, ah1, at1, ehb, etb, H, D);
    fold_we_kernel<<<1, 256, 0, stream>>>(We1, ae1, ewb, H, D);
    init_ms_kernel<<<cdiv((long)NNODES * H, 256), 256, 0, stream>>>(
        maxb, sumb, NNODES * H);
    edge_logits_kernel<<<cdiv(NEDGES, 256), 256, 0, stream>>>(
        eattr, src, dst, ewb, ehb, etb, exb, maxb, H);
    edge_exp_sum_kernel<<<cdiv((long)NEDGES * H, 256), 256, 0, stream>>>(
        dst, maxb, exb, sumb, H);
    init_out_kernel<<<cdiv((long)NNODES * HD, 256), 256, 0, stream>>>(
        out, b1, HD, (long)NNODES * HD);
    edge_agg_kernel<<<cdiv((long)NEDGES * HD, 256), 256, 0, stream>>>(
        src, dst, feat1, exb, sumb, out, H, D);
    // relu_out=False for layer 1: done.
  }
  (void)in_sizes; (void)n_in; (void)out_size; (void)ws_size;
}